// Aggregator_2422361555371
// MI455X (gfx1250) — compile-verified
//
#include <hip/hip_runtime.h>

// CDNA5 / gfx1250: wave32, WMMA f32 16x16x4.
typedef float v2f __attribute__((ext_vector_type(2)));
typedef float v8f __attribute__((ext_vector_type(8)));

#define WAVES_PER_BLOCK 8   // 256 threads = 8 wave32

// One wave per (b, iter) unit.
//   scores[32]   = (1/64) * nr(32x64) . ue(64)        -- via V_WMMA_F32_16X16X4_F32
//   w            = softmax(scores)
//   out[64]      = relu( (1/32) * sum_n w[n]*nv[n,:] ) -- coalesced VALU pass
__global__ __launch_bounds__(256) void aggregator_wmma_kernel(
    const float* __restrict__ nv,   // [B, 8*32, 64] neighbor_vectors
    const float* __restrict__ nr,   // [B, 8*32, 64] neighbor_relations
    const float* __restrict__ ue,   // [B, 64]       user_embeddings
    float* __restrict__ out)        // [B, 8, 64]
{
    __shared__ float sScores[WAVES_PER_BLOCK * 32];

    const int lane = threadIdx.x & 31;
    const int wave = threadIdx.x >> 5;
    const int u    = blockIdx.x * WAVES_PER_BLOCK + wave;   // (b*8 + iter)

    const float* __restrict__ nrb = nr + (size_t)u * 2048;      // 32x64 tile
    const float* __restrict__ nvb = nv + (size_t)u * 2048;      // 32x64 tile
    const float* __restrict__ ueb = ue + (size_t)(u >> 3) * 64; // per-b vector

    // Prefetch the nv tile (8 KB) so it's in flight while WMMA stage runs.
    __builtin_prefetch(nvb + lane * 64, 0, 0);
    __builtin_prefetch(nvb + lane * 64 + 32, 0, 0);

    // ---- Stage 1: scores via WMMA f32 16x16x4 ------------------------------
    // A layout (ISA 7.12.2, 32-bit A 16x4): lanes 0-15 hold {K=0,K=1} in V0/V1,
    // lanes 16-31 hold {K=2,K=3}.  B uses the same K striping across lane
    // halves; we replicate ue across all 16 N columns so every column of D
    // equals the score vector.
    const int koff = (lane >> 4) << 1;   // 0 for lanes 0-15, 2 for lanes 16-31
    const int mrow = lane & 15;

    v8f c0 = {};   // neighbors 0..15
    v8f c1 = {};   // neighbors 16..31
#pragma unroll
    for (int k0 = 0; k0 < 64; k0 += 4) {
        v2f bfrag = *(const v2f*)(ueb + k0 + koff);
        v2f a0    = *(const v2f*)(nrb + mrow * 64        + k0 + koff);
        v2f a1    = *(const v2f*)(nrb + (mrow + 16) * 64 + k0 + koff);
        c0 = __builtin_amdgcn_wmma_f32_16x16x4_f32(false, a0, false, bfrag,
                                                   (short)0, c0, false, false);
        c1 = __builtin_amdgcn_wmma_f32_16x16x4_f32(false, a1, false, bfrag,
                                                   (short)0, c1, false, false);
    }

    // D layout: VGPR r, lanes 0-15 -> (M=r, N=lane); lanes 16-31 -> (M=8+r).
    // All N columns identical, so lane 0 / lane 16 hold the full score set.
    float* s = sScores + wave * 32;
    if (lane == 0) {
#pragma unroll
        for (int r = 0; r < 8; ++r) { s[r]      = c0[r]; s[16 + r] = c1[r]; }
    } else if (lane == 16) {
#pragma unroll
        for (int r = 0; r < 8; ++r) { s[8 + r]  = c0[r]; s[24 + r] = c1[r]; }
    }
    __syncthreads();

    // ---- Stage 2: softmax over the 32 neighbors (one score per lane) -------
    float sc = s[lane] * (1.0f / 64.0f);   // mean over dim

    float mx = sc;
#pragma unroll
    for (int off = 16; off >= 1; off >>= 1)
        mx = fmaxf(mx, __shfl_xor(mx, off, 32));
    float e = __expf(sc - mx);
    float sum = e;
#pragma unroll
    for (int off = 16; off >= 1; off >>= 1)
        sum += __shfl_xor(sum, off, 32);
    float w = e / sum;

    // ---- Stage 3: weighted neighbor aggregation, coalesced -----------------
    // Lane owns dims [2*lane, 2*lane+1]; each neighbor row read as one
    // contiguous 256B wave transaction (global_load_b64 per lane).
    v2f acc = {};
#pragma unroll 8
    for (int n = 0; n < 32; ++n) {
        float wn = __shfl(w, n, 32);
        v2f row  = *(const v2f*)(nvb + n * 64 + 2 * lane);
        acc.x = fmaf(wn, row.x, acc.x);
        acc.y = fmaf(wn, row.y, acc.y);
    }

    v2f o;
    o.x = fmaxf(acc.x * (1.0f / 32.0f), 0.0f);   // mean over nsize + relu
    o.y = fmaxf(acc.y * (1.0f / 32.0f), 0.0f);
    *(v2f*)(out + (size_t)u * 64 + 2 * lane) = o;
}

extern "C" void kernel_launch(void* const* d_in, const int* in_sizes, int n_in,
                              void* d_out, int out_size, void* d_ws, size_t ws_size,
                              hipStream_t stream) {
    // setup_inputs order: 0 self_vectors (unused), 1 neighbor_vectors,
    // 2 neighbor_relations, 3 user_embeddings, 4 neighbor_size (scalar).
    const float* nv  = (const float*)d_in[1];
    const float* nr  = (const float*)d_in[2];
    const float* ue  = (const float*)d_in[3];
    float*       out = (float*)d_out;

    // self_vectors is [B, 8, 64] -> units = B*8, always a multiple of 8.
    const int units  = in_sizes[0] / 64;
    const int blocks = units / WAVES_PER_BLOCK;

    aggregator_wmma_kernel<<<blocks, 256, 0, stream>>>(nv, nr, ue, out);
}